// Detect_90288802497002
// MI455X (gfx1250) — compile-verified
//
#include <hip/hip_runtime.h>
#include <math.h>

// ---- problem constants (match reference setup_inputs) ----
#define BATCH       32
#define NPRIORS     24564
#define NUM_CLASSES 81
#define TOPK        200
#define CONF_T      0.01f
#define NMS_T       0.45f
#define VAR0        0.1f
#define VAR1        0.2f

#define NTHREADS    256
#define PADP        24576   // NPRIORS padded up to multiple of NTHREADS (96*256)
#define NWAVES      (NTHREADS / 32)

typedef __attribute__((ext_vector_type(4))) unsigned int v4u;
typedef __attribute__((ext_vector_type(8))) int          v8i;
typedef __attribute__((ext_vector_type(4))) int          v4i;

__device__ __forceinline__ void argmax_combine(float& v, int& i, float v2, int i2) {
    // larger value wins; on tie, smaller index wins (jax.lax.top_k stability)
    if (v2 > v || (v2 == v && i2 < i)) { v = v2; i = i2; }
}

__global__ __launch_bounds__(NTHREADS)
void detect_kernel(const float* __restrict__ loc,
                   const float* __restrict__ conf,
                   const float* __restrict__ prior,
                   float* __restrict__ out)
{
    __shared__ float s_scores[PADP];            // 98304 B: class column staged in LDS
    __shared__ float s_sv[NTHREADS];
    __shared__ int   s_si[NTHREADS];
    __shared__ float s_wv[NWAVES];
    __shared__ int   s_wi[NWAVES];
    __shared__ float s_bestv;
    __shared__ int   s_besti;
    __shared__ float s_topv[TOPK];
    __shared__ int   s_topi[TOPK];
    __shared__ __align__(16) float s_loc[TOPK * 4];   // candidate loc rows (async b128)
    __shared__ __align__(16) float s_pri[TOPK * 4];   // candidate prior rows (async b128)
    __shared__ float s_x1[TOPK], s_y1[TOPK], s_x2[TOPK], s_y2[TOPK], s_area[TOPK];
    __shared__ int   s_vld[TOPK], s_sup[TOPK], s_keep[TOPK], s_tgt[TOPK];
    __shared__ float s_out[TOPK * 5];

    const int tid  = threadIdx.x;
    const int lane = tid & 31;
    const int wid  = tid >> 5;
    const int b    = blockIdx.x / NUM_CLASSES;
    const int c    = blockIdx.x % NUM_CLASSES;

    float* outp = out + (size_t)(b * NUM_CLASSES + c) * (TOPK * 5);

    // background class is zeroed by the reference (out.at[:,0].set(0))
    if (c == 0) {
        for (int i = tid; i < TOPK * 5; i += NTHREADS) outp[i] = 0.0f;
        return;
    }

    // ---- Phase 1: stage the strided class column conf[b, :, c] into LDS ----
    // The column is a (24564 x 1) tile with a 324 B row stride: exactly the TDM
    // descriptor model. One tensor_load_to_lds (TENSORcnt) replaces ~768 per-wave
    // async loads and offloads the gather to the DMA engine.
#if __has_builtin(__builtin_amdgcn_tensor_load_to_lds)
    {
        if (tid < 32) {   // wave 0 issues the DMA once
            unsigned long long ga = (unsigned long long)(uintptr_t)
                (conf + (size_t)b * NPRIORS * NUM_CLASSES + c);
            // generic flat LDS pointers keep the workgroup-relative byte offset
            // in the low 32 bits (aperture lives in the high bits)
            unsigned int lds_base = (unsigned int)(uintptr_t)(&s_scores[0]);

            v4u g0;
            g0.x = 0x1u;                                   // count=1, is_restore=0, no gather
            g0.y = lds_base;                               // lds_addr (bytes)
            g0.z = (unsigned int)(ga & 0xFFFFFFFFull);     // global_addr[31:0]
            g0.w = (unsigned int)((ga >> 32) & 0x1FFFFFFull)
                 | (2u << 30);                             // global_addr[56:32] | type=2
            v8i g1;
            g1[0] = (int)(2u << 16);                       // workgroup_mask=0, data_size=2 (4B)
            g1[1] = (int)(1u << 16);                       // tensor_dim0 = 1   (bits 48..79)
            g1[2] = (int)((unsigned)(NPRIORS & 0xFFFF) << 16); // tensor_dim1[15:0] (bits 80..95)
            g1[3] = (int)(1u << 16);                       // tensor_dim1 hi=0, tile_dim0=1
            g1[4] = NPRIORS;                               // tile_dim1 = 24564, tile_dim2 = 0
            g1[5] = NUM_CLASSES;                           // tensor_dim0_stride = 81 (low 32)
            g1[6] = 0;                                     // stride hi = 0, dim1_stride lo = 0
            g1[7] = 0;                                     // dim1_stride hi = 0 (dim2 unused)
            v4i g2 = {0, 0, 0, 0};                         // 2D tensor: groups 2/3 unused
            v4i g3 = {0, 0, 0, 0};
            v8i g4 = {0, 0, 0, 0, 0, 0, 0, 0};             // surplus descriptor space
                                                           // (6-arg toolchain form); count=1
                                                           // so it is ignored
            __builtin_amdgcn_tensor_load_to_lds(g0, g1, g2, g3, g4, 0);
            __builtin_amdgcn_s_wait_tensorcnt(0);
        }
        for (int i = NPRIORS + tid; i < PADP; i += NTHREADS)
            s_scores[i] = -INFINITY;                       // pad = below-threshold
        __syncthreads();
    }
#else
    {
        const float* col = conf + (size_t)b * NPRIORS * NUM_CLASSES + c;
        for (int i = tid; i < PADP; i += NTHREADS) {
            if (i < NPRIORS) {
                unsigned long long ga =
                    (unsigned long long)(uintptr_t)(col + (size_t)i * NUM_CLASSES);
                unsigned int la = (unsigned int)(uintptr_t)(&s_scores[i]);
                asm volatile("global_load_async_to_lds_b32 %0, %1, off"
                             :: "v"(la), "v"(ga) : "memory");
            } else {
                s_scores[i] = -INFINITY;
            }
        }
        asm volatile("s_wait_asynccnt 0" ::: "memory");
        __syncthreads();
    }
#endif

    // ---- Phase 2: initial per-thread strip argmax (strided strips: bank-conflict-free)
    {
        float bv = -INFINITY; int bi = 0x7FFFFFFF;
        for (int j = tid; j < PADP; j += NTHREADS) {
            float s   = s_scores[j];
            float key = (s > CONF_T) ? s : -INFINITY;      // masked score
            argmax_combine(bv, bi, key, j);
        }
        s_sv[tid] = bv; s_si[tid] = bi;
    }
    __syncthreads();

    // ---- Phase 3: tournament top-K selection (stable, descending) ----
    for (int k = 0; k < TOPK; ++k) {
        float v  = s_sv[tid];
        int   ix = s_si[tid];
        #pragma unroll
        for (int off = 16; off > 0; off >>= 1) {           // wave32 reduction
            float v2 = __shfl_down(v, off, 32);
            int   i2 = __shfl_down(ix, off, 32);
            argmax_combine(v, ix, v2, i2);
        }
        if (lane == 0) { s_wv[wid] = v; s_wi[wid] = ix; }
        __syncthreads();
        if (wid == 0) {
            float v2 = (lane < NWAVES) ? s_wv[lane] : -INFINITY;
            int   i2 = (lane < NWAVES) ? s_wi[lane] : 0x7FFFFFFF;
            #pragma unroll
            for (int off = 4; off > 0; off >>= 1) {
                float v3 = __shfl_down(v2, off, 32);
                int   i3 = __shfl_down(i2, off, 32);
                argmax_combine(v2, i2, v3, i3);
            }
            if (lane == 0) { s_bestv = v2; s_besti = i2; }
        }
        __syncthreads();
        float bv = s_bestv;
        int   bi = s_besti;
        if (bv == -INFINITY) {
            // nothing above threshold remains: remaining slots are invalid
            for (int j = k + tid; j < TOPK; j += NTHREADS) { s_topv[j] = -INFINITY; s_topi[j] = 0; }
            __syncthreads();
            break;
        }
        if (tid == 0) { s_topv[k] = bv; s_topi[k] = bi; }
        if (tid == (bi & (NTHREADS - 1))) {                // strip owner: mark + rescan
            s_scores[bi] = -INFINITY;
            float nbv = -INFINITY; int nbi = 0x7FFFFFFF;
            for (int j = tid; j < PADP; j += NTHREADS) {
                float s   = s_scores[j];
                float key = (s > CONF_T) ? s : -INFINITY;
                argmax_combine(nbv, nbi, key, j);
            }
            s_sv[tid] = nbv; s_si[tid] = nbi;
        }
        __syncthreads();
    }
    __syncthreads();

    // ---- Phase 4: async-gather the 200 candidate loc/prior rows, then decode ----
    int   vld = 0;
    float tv  = 0.0f;
    if (tid < TOPK) {
        tv  = s_topv[tid];
        vld = (tv != -INFINITY) ? 1 : 0;                   // valid = isfinite(masked score)
        if (vld) {
            int bi = s_topi[tid];
            unsigned long long gal = (unsigned long long)(uintptr_t)
                (loc + ((size_t)b * NPRIORS + bi) * 4);
            unsigned long long gap = (unsigned long long)(uintptr_t)(prior + (size_t)bi * 4);
            unsigned int lal = (unsigned int)(uintptr_t)(&s_loc[tid * 4]);
            unsigned int lap = (unsigned int)(uintptr_t)(&s_pri[tid * 4]);
            asm volatile("global_load_async_to_lds_b128 %0, %1, off"
                         :: "v"(lal), "v"(gal) : "memory");
            asm volatile("global_load_async_to_lds_b128 %0, %1, off"
                         :: "v"(lap), "v"(gap) : "memory");
        }
    }
    asm volatile("s_wait_asynccnt 0" ::: "memory");        // per-wave: own loads done
    if (tid < TOPK) {
        float x1 = 0.f, y1 = 0.f, x2 = 0.f, y2 = 0.f, ar = 0.f;
        if (vld) {
            float4 l4 = reinterpret_cast<float4*>(s_loc)[tid];
            float4 p4 = reinterpret_cast<float4*>(s_pri)[tid];
            float cx = p4.x + l4.x * VAR0 * p4.z;
            float cy = p4.y + l4.y * VAR0 * p4.w;
            float w  = p4.z * expf(l4.z * VAR1);
            float h  = p4.w * expf(l4.w * VAR1);
            x1 = cx - w * 0.5f;
            y1 = cy - h * 0.5f;
            x2 = x1 + w;
            y2 = y1 + h;
            ar = (x2 - x1) * (y2 - y1);
        }
        s_x1[tid] = x1; s_y1[tid] = y1; s_x2[tid] = x2; s_y2[tid] = y2; s_area[tid] = ar;
        s_vld[tid] = vld; s_sup[tid] = 0;
    }
    __syncthreads();

    // ---- Phase 5: greedy NMS, identical order to the reference scan ----
    float x1j = 0.f, y1j = 0.f, x2j = 0.f, y2j = 0.f, aj = 0.f;
    if (tid < TOPK) { x1j = s_x1[tid]; y1j = s_y1[tid]; x2j = s_x2[tid]; y2j = s_y2[tid]; aj = s_area[tid]; }
    for (int i = 0; i < TOPK; ++i) {
        int kept = s_vld[i] && !s_sup[i];
        if (tid == 0) s_keep[i] = kept;
        if (kept && tid < TOPK && tid > i) {
            float iw = fmaxf(fminf(s_x2[i], x2j) - fmaxf(s_x1[i], x1j), 0.0f);
            float ih = fmaxf(fminf(s_y2[i], y2j) - fmaxf(s_y1[i], y1j), 0.0f);
            float inter = iw * ih;
            float uni   = s_area[i] + aj - inter;
            float iou   = (uni > 0.0f) ? (inter / uni) : 0.0f;
            if (iou > NMS_T) s_sup[tid] = 1;   // only thread tid writes s_sup[tid]
        }
        __syncthreads();
    }

    // ---- Phase 6: compaction (cumsum(keep)-1 scatter) + coalesced store ----
    if (tid == 0) {
        int r = 0;
        for (int i = 0; i < TOPK; ++i) s_tgt[i] = s_keep[i] ? (r++) : TOPK;
    }
    for (int i = tid; i < TOPK * 5; i += NTHREADS) s_out[i] = 0.0f;
    __syncthreads();
    if (tid < TOPK && s_keep[tid]) {
        int t = s_tgt[tid];                    // < TOPK whenever kept
        s_out[t * 5 + 0] = tv;                 // kept -> valid -> finite masked score
        s_out[t * 5 + 1] = x1j;
        s_out[t * 5 + 2] = y1j;
        s_out[t * 5 + 3] = x2j;
        s_out[t * 5 + 4] = y2j;
    }
    __syncthreads();
    for (int i = tid; i < TOPK * 5; i += NTHREADS) outp[i] = s_out[i];
}

extern "C" void kernel_launch(void* const* d_in, const int* in_sizes, int n_in,
                              void* d_out, int out_size, void* d_ws, size_t ws_size,
                              hipStream_t stream) {
    const float* loc   = (const float*)d_in[0];   // (B, P, 4)
    const float* conf  = (const float*)d_in[1];   // (B, P, C)
    const float* prior = (const float*)d_in[2];   // (P, 4)
    float* out = (float*)d_out;                   // (B, C, K, 5)
    dim3 grid(BATCH * NUM_CLASSES);
    detect_kernel<<<grid, NTHREADS, 0, stream>>>(loc, conf, prior, out);
}